// FastSpectralAttention_39273180954790
// MI455X (gfx1250) — compile-verified
//
#include <hip/hip_runtime.h>

// ---------------------------------------------------------------------------
// FastSpectralAttention for MI455X (gfx1250, wave32, WMMA f32 16x16x4)
//
// B=2, T=4096, E=1024, H=8, K(eig)=32. SCALE = 32^-0.5.
// Pipeline:
//   k_qkv   : qkv = x @ Wqkv, fused basis-multiply + scatter to q/k/v [bh,t,e]
//   k_state : per 64-token chunk, S_chunk = K_chunk^T @ V_chunk (32x32)
//   k_scan  : exclusive prefix-sum of chunk states along chunks (per b,h)
//   k_attn  : attn_chunk = (tril(Q K^T) @ V + Q @ S_prev) * SCALE
//   k_out   : out = attn @ Wout + bout
// ---------------------------------------------------------------------------

typedef __attribute__((ext_vector_type(2))) float v2f;
typedef __attribute__((ext_vector_type(8))) float v8f;

#define N_B     2
#define N_T     4096
#define N_E     1024
#define N_H     8
#define N_K     32
#define CQKV    768        // 3*H*K
#define CHK     64         // chunk length (tokens)
#define NCHUNK  64         // T / CHK
#define NBH     16         // B*H
#define SCALE_F 0.17677669529663687f

__device__ __forceinline__ v8f wmma4(v2f a, v2f b, v8f c) {
  // D = A(16x4 f32) * B(4x16 f32) + C(16x16 f32)
  return __builtin_amdgcn_wmma_f32_16x16x4_f32(
      /*neg_a=*/false, a, /*neg_b=*/false, b,
      /*c_mod=*/(short)0, c, /*reuse_a=*/false, /*reuse_b=*/false);
}

__device__ __forceinline__ int waveid() {
  return __builtin_amdgcn_readfirstlane((int)(threadIdx.x >> 5));
}

// A-fragment from row-major matrix: A[m, kk], lane = m + 16*(kk>>1), 2 consecutive k's.
__device__ __forceinline__ v2f loadA(const float* __restrict__ p, int ld,
                                     int row0, int k0, int lane) {
  int m = lane & 15, h = lane >> 4;
  const float* q = p + (long)(row0 + m) * ld + k0 + 2 * h;
  v2f r = { q[0], q[1] };
  return r;
}

// B-fragment from row-major matrix: B[kk, n], lane = n + 16*(kk>>1), 2 rows (stride ld).
__device__ __forceinline__ v2f loadB(const float* __restrict__ p, int ld,
                                     int k0, int col0, int lane) {
  int n = lane & 15, h = lane >> 4;
  const float* q = p + (long)(k0 + 2 * h) * ld + col0 + n;
  v2f r = { q[0], q[ld] };
  return r;
}

// ---------------------------------------------------------------------------
// Kernel 1: qkv = x @ Wqkv ; scatter to q/k/v [bh][t][e], basis applied to q,k.
// grid = 3072 blocks * 128 thr (4 waves); wave -> one 16x32 tile of (8192 x 768).
// The 32-wide N tile == one (which,h) block, so the scatter is wave-uniform.
// ---------------------------------------------------------------------------
__global__ void k_qkv(const float* __restrict__ x, const float* __restrict__ W,
                      const float* __restrict__ basis,
                      float* __restrict__ qw, float* __restrict__ kw,
                      float* __restrict__ vw) {
  int lane = threadIdx.x & 31;
  int tile = blockIdx.x * 4 + waveid();      // 512 * 24 tiles (16 x 32)
  int mt = tile / 24, nb = tile % 24;        // scalar
  int row0 = mt * 16, col0 = nb * 32;

  v8f acc0 = {}, acc1 = {};
#pragma unroll 4
  for (int k0 = 0; k0 < N_E; k0 += 4) {
    v2f a  = loadA(x, N_E, row0, k0, lane);          // shared by both N-halves
    v2f b0 = loadB(W, CQKV, k0, col0, lane);
    v2f b1 = loadB(W, CQKV, k0, col0 + 16, lane);
    acc0 = wmma4(a, b0, acc0);
    acc1 = wmma4(a, b1, acc1);
  }

  int which = col0 >> 8;                     // 0=q 1=k 2=v (scalar)
  int h     = (col0 & 255) >> 5;             // scalar
  int bb    = row0 >> 12;                    // scalar (4096 % 16 == 0)
  int t0    = (row0 & 4095) + 8 * (lane >> 4);
  int e0    = lane & 15, e1 = e0 + 16;
  float* dst = (which == 0) ? qw : (which == 1) ? kw : vw;
  long obase = (((long)(bb * N_H + h) * N_T) + t0) * N_K;
  if (which != 2) {
    const float* bp = basis + ((long)(bb * N_T + t0)) * N_K;
#pragma unroll
    for (int r = 0; r < 8; ++r) {
      float s0 = bp[r * N_K + e0], s1 = bp[r * N_K + e1];
      dst[obase + r * N_K + e0] = acc0[r] * s0;
      dst[obase + r * N_K + e1] = acc1[r] * s1;
    }
  } else {
#pragma unroll
    for (int r = 0; r < 8; ++r) {
      dst[obase + r * N_K + e0] = acc0[r];
      dst[obase + r * N_K + e1] = acc1[r];
    }
  }
}

// ---------------------------------------------------------------------------
// Kernel 2a: per-chunk state S = K_chunk^T @ V_chunk (32x32, Kdim = 64).
// grid = 1024 blocks (one per (bh,chunk)) * 128 thr; wave -> one 16x16 tile.
// ---------------------------------------------------------------------------
__global__ void k_state(const float* __restrict__ kw, const float* __restrict__ vw,
                        float* __restrict__ sw) {
  int lane = threadIdx.x & 31, wave = waveid();
  int chunk = blockIdx.x;                    // bh*64 + c
  int bh = chunk >> 6, c = chunk & 63;
  const float* kb = kw + ((long)bh * N_T + c * CHK) * N_K;  // 64 x 32 row-major
  const float* vb = vw + ((long)bh * N_T + c * CHK) * N_K;
  int ti = wave >> 1, tj = wave & 1;         // scalar

  v8f acc = {};
  int m = lane & 15, hh = lane >> 4;
#pragma unroll 4
  for (int k0 = 0; k0 < CHK; k0 += 4) {
    // A = K^T: A[m,kk] = kchunk[kk][ti*16+m]  (two kk's, stride 32 floats)
    const float* pa = kb + (k0 + 2 * hh) * N_K + ti * 16 + m;
    v2f a = { pa[0], pa[N_K] };
    // B = V:   B[kk,n] = vchunk[kk][tj*16+n]
    const float* pb = vb + (k0 + 2 * hh) * N_K + tj * 16 + m;
    v2f b = { pb[0], pb[N_K] };
    acc = wmma4(a, b, acc);
  }
  float* dst = sw + (long)chunk * (N_K * N_K);
  int mh = 8 * (lane >> 4);
#pragma unroll
  for (int r = 0; r < 8; ++r)
    dst[(ti * 16 + mh + r) * N_K + tj * 16 + m] = acc[r];
}

// ---------------------------------------------------------------------------
// Kernel 2b: in-place exclusive scan of chunk states along chunk axis.
// grid = 16 blocks (one per bh) * 256 thr; each thread owns 4 of 1024 elems.
// ---------------------------------------------------------------------------
__global__ void k_scan(float* __restrict__ sw) {
  int bh = blockIdx.x;
  int idx = threadIdx.x * 4;
  float run0 = 0.f, run1 = 0.f, run2 = 0.f, run3 = 0.f;
  for (int c = 0; c < NCHUNK; ++c) {
    float4* p = (float4*)(sw + ((long)(bh * NCHUNK + c)) * (N_K * N_K) + idx);
    float4 v = *p;
    float4 o = { run0, run1, run2, run3 };
    *p = o;
    run0 += v.x; run1 += v.y; run2 += v.z; run3 += v.w;
  }
}

// ---------------------------------------------------------------------------
// Kernel 3: per-chunk attention.
//   A = tril(Q K^T) (64x64, masked, staged via LDS for layout conversion)
//   attn = (A @ V + Q @ S_prev) * SCALE   -> attn layout [b][t][h*32+e]
// grid = 1024 blocks * 256 thr (8 waves).
// ---------------------------------------------------------------------------
#define ALD 65   // LDS row stride (floats) to dodge bank conflicts
__global__ void k_attn(const float* __restrict__ qw, const float* __restrict__ kw,
                       const float* __restrict__ vw, const float* __restrict__ sw,
                       float* __restrict__ attn) {
  __shared__ float a_lds[CHK * ALD];
  int lane = threadIdx.x & 31, wave = waveid();
  int chunk = blockIdx.x;
  int bh = chunk >> 6, c = chunk & 63;
  int bb = bh >> 3, h = bh & 7;
  const float* qb = qw + ((long)bh * N_T + c * CHK) * N_K;
  const float* kb = kw + ((long)bh * N_T + c * CHK) * N_K;
  const float* vb = vw + ((long)bh * N_T + c * CHK) * N_K;
  const float* sb = sw + (long)chunk * (N_K * N_K);   // exclusive prefix state

  // ---- Phase 1: A = tril(Q K^T); 4x4 tiles of 16x16, wave -> 2 tiles ----
#pragma unroll
  for (int u = 0; u < 2; ++u) {
    int tt = wave * 2 + u;                   // scalar
    int ti = tt >> 2, tj = tt & 3;           // scalar
    v8f acc = {};
    if (ti >= tj) {                          // scalar branch: EXEC stays all-1s
#pragma unroll
      for (int k0 = 0; k0 < N_K; k0 += 4) {
        v2f a = loadA(qb, N_K, ti * 16, k0, lane);
        // B[e,j] = k[tj*16+j][e]: two consecutive e's -> one b64 load
        int n = lane & 15, hh = lane >> 4;
        const float* pb = kb + (tj * 16 + n) * N_K + k0 + 2 * hh;
        v2f b = { pb[0], pb[1] };
        acc = wmma4(a, b, acc);
      }
    }
    int n = lane & 15, mh = 8 * (lane >> 4);
#pragma unroll
    for (int r = 0; r < 8; ++r) {
      int gi = ti * 16 + mh + r, gj = tj * 16 + n;
      a_lds[gi * ALD + gj] = (gj <= gi) ? acc[r] : 0.0f;   // inclusive causal
    }
  }
  __syncthreads();

  // ---- Phase 2: out tile (4 row-tiles x 2 col-tiles) = A@V + Q@S_prev ----
  int ti = wave >> 1, cj = wave & 1;         // scalar
  v8f acc = {};
  int m = lane & 15, hh = lane >> 4;
#pragma unroll 4
  for (int k0 = 0; k0 < CHK; k0 += 4) {      // A @ V, Kdim = 64
    const float* pa = a_lds + (ti * 16 + m) * ALD + k0 + 2 * hh;
    v2f a = { pa[0], pa[1] };
    const float* pb = vb + (k0 + 2 * hh) * N_K + cj * 16 + m;
    v2f b = { pb[0], pb[N_K] };
    acc = wmma4(a, b, acc);
  }
#pragma unroll
  for (int k0 = 0; k0 < N_K; k0 += 4) {      // Q @ S_prev, Kdim = 32
    v2f a = loadA(qb, N_K, ti * 16, k0, lane);
    const float* pb = sb + (k0 + 2 * hh) * N_K + cj * 16 + m;
    v2f b = { pb[0], pb[N_K] };
    acc = wmma4(a, b, acc);
  }
  int e  = cj * 16 + m;
  int mh = 8 * (lane >> 4);
#pragma unroll
  for (int r = 0; r < 8; ++r) {
    int t = c * CHK + ti * 16 + mh + r;
    attn[((long)(bb * N_T + t)) * (N_H * N_K) + h * N_K + e] = acc[r] * SCALE_F;
  }
}

// ---------------------------------------------------------------------------
// Kernel 4: out = attn @ Wout + bout   (8192 x 256 x 1024)
// grid = 4096 blocks * 128 thr (4 waves); wave -> one 16x32 tile (shared A).
// ---------------------------------------------------------------------------
__global__ void k_out(const float* __restrict__ attn, const float* __restrict__ Wout,
                      const float* __restrict__ bout, float* __restrict__ out) {
  int lane = threadIdx.x & 31;
  int tile = blockIdx.x * 4 + waveid();      // 512 * 32 tiles (16 x 32)
  int mt = tile >> 5, nb = tile & 31;        // scalar
  int row0 = mt * 16, col0 = nb * 32;

  v8f acc0 = {}, acc1 = {};
#pragma unroll 4
  for (int k0 = 0; k0 < N_H * N_K; k0 += 4) {
    v2f a  = loadA(attn, N_H * N_K, row0, k0, lane); // shared by both N-halves
    v2f b0 = loadB(Wout, N_E, k0, col0, lane);
    v2f b1 = loadB(Wout, N_E, k0, col0 + 16, lane);
    acc0 = wmma4(a, b0, acc0);
    acc1 = wmma4(a, b1, acc1);
  }
  int n0 = col0 + (lane & 15), n1 = n0 + 16;
  float bias0 = bout[n0], bias1 = bout[n1];
  int mbase = row0 + 8 * (lane >> 4);
#pragma unroll
  for (int r = 0; r < 8; ++r) {
    out[(long)(mbase + r) * N_E + n0] = acc0[r] + bias0;
    out[(long)(mbase + r) * N_E + n1] = acc1[r] + bias1;
  }
}

// ---------------------------------------------------------------------------
// Launch
// ---------------------------------------------------------------------------
extern "C" void kernel_launch(void* const* d_in, const int* in_sizes, int n_in,
                              void* d_out, int out_size, void* d_ws, size_t ws_size,
                              hipStream_t stream) {
  (void)in_sizes; (void)n_in; (void)out_size; (void)ws_size;
  const float* x     = (const float*)d_in[0];   // (2,4096,1024)
  const float* basis = (const float*)d_in[1];   // (2,4096,32)
  // d_in[2] = spectral_freqs: unused by the reference computation
  const float* Wqkv  = (const float*)d_in[3];   // (1024,768)
  const float* Wout  = (const float*)d_in[4];   // (256,1024)
  const float* bout  = (const float*)d_in[5];   // (1024,)
  float* out = (float*)d_out;

  float* ws = (float*)d_ws;
  const long QKV = (long)NBH * N_T * N_K;       // 2,097,152 floats each
  float* qw   = ws;
  float* kw   = ws + QKV;
  float* vw   = ws + 2 * QKV;
  float* attn = ws + 3 * QKV;                   // (8192 x 256)
  float* sw   = ws + 4 * QKV;                   // 1024 states x 32x32

  k_qkv  <<<3072, 128, 0, stream>>>(x, Wqkv, basis, qw, kw, vw);
  k_state<<<1024, 128, 0, stream>>>(kw, vw, sw);
  k_scan <<<  16, 256, 0, stream>>>(sw);
  k_attn <<<1024, 256, 0, stream>>>(qw, kw, vw, sw, attn);
  k_out  <<<4096, 128, 0, stream>>>(attn, Wout, bout, out);
}